// DiagonalColCausalLinear_12481174962953
// MI455X (gfx1250) — compile-verified
//
#include <hip/hip_runtime.h>

// V_WMMA_F32_16X16X4_F32: D(16x16,f32) = A(16x4,f32) x B(4x16,f32) + C
typedef __attribute__((ext_vector_type(2))) float v2f;
typedef __attribute__((ext_vector_type(8))) float v8f;

// out[r,t] = x[r,t]*v2[t] + bias[t] + sum_{s<t} x[r,s]*v[s]*d^(t-s)
// Two-pass segmented chunked scan:
//   Pass 1: per (16-row tile, segment) -> L[row,g] = sum_{s in g} x*v*d^(segEnd-s)
//   Pass 2: carry_in(g) = Horner(L[row,0..g-1], d^SEG), then per 16-col chunk
//           compute the 16x16 diagonal block of M via 4x WMMA f32 16x16x4.

__global__ __launch_bounds__(256) void seg_carry_kernel(
    const float* __restrict__ x, const float* __restrict__ w,
    const float* __restrict__ decay, float* __restrict__ L,
    int S, int R, int G, int SEG)
{
    const int lane = threadIdx.x & 31;
    const int task = blockIdx.x * (blockDim.x >> 5) + (threadIdx.x >> 5);
    const int tiles = R >> 4;
    const int rowTile = task / G, g = task - rowTile * G;
    if (rowTile >= tiles) return;                 // wave-uniform

    const int h = lane >> 4, u = lane & 15;
    const float d = fminf(fmaxf(decay[1], 0.9f), 1.0f);

    float dw[4][2];                               // d^(16-j), j = 4s+2h+r
    #pragma unroll
    for (int s = 0; s < 4; ++s)
        #pragma unroll
        for (int r = 0; r < 2; ++r)
            dw[s][r] = __powf(d, (float)(16 - (4*s + 2*h + r)));
    const float d16 = __powf(d, 16.0f);

    const float* xrow = x + (size_t)(rowTile*16 + u) * (size_t)S + g * SEG;
    const float* wseg = w + g * SEG;

    // Per-lane half-recurrence; halves combine linearly at the end.
    float Ls = 0.0f;
    for (int cb = 0; cb < SEG; cb += 16) {
        float partial = 0.0f;
        #pragma unroll
        for (int s = 0; s < 4; ++s) {
            const int col = cb + 4*s + 2*h;
            const v2f a  = *(const v2f*)(xrow + col);
            const v2f vv = *(const v2f*)(wseg + col);
            partial += a.x*vv.x*dw[s][0] + a.y*vv.y*dw[s][1];
        }
        Ls = d16 * Ls + partial;
    }
    Ls += __shfl_xor(Ls, 16, 32);
    if (lane < 16) L[(size_t)(rowTile*16 + u) * G + g] = Ls;
}

__global__ __launch_bounds__(256) void seg_scan_kernel(
    const float* __restrict__ x, const float* __restrict__ w,
    const float* __restrict__ wd, const float* __restrict__ bias,
    const float* __restrict__ decay, const float* __restrict__ L,
    float* __restrict__ out, int S, int R, int G, int SEG)
{
    const int lane = threadIdx.x & 31;
    const int task = blockIdx.x * (blockDim.x >> 5) + (threadIdx.x >> 5);
    const int tiles = R >> 4;
    const int rowTile = task / G, g = task - rowTile * G;
    if (rowTile >= tiles) return;                 // wave-uniform; EXEC all-ones
    const int rowBase = rowTile * 16;

    const int h = lane >> 4, u = lane & 15;
    const float d = fminf(fmaxf(decay[1], 0.9f), 1.0f);

    float tri[4][2];   // (u>j) ? d^(u-j) : 0   (strict upper triangle)
    float dw [4][2];   // d^(16-j)              (carry-update weights)
    bool  dg [4][2];   // u == j                (diagonal)
    #pragma unroll
    for (int s = 0; s < 4; ++s) {
        #pragma unroll
        for (int r = 0; r < 2; ++r) {
            const int j = 4*s + 2*h + r;
            tri[s][r] = (u > j) ? __powf(d, (float)(u - j)) : 0.0f;
            dw [s][r] = __powf(d, (float)(16 - j));
            dg [s][r] = (u == j);
        }
    }
    const float dpowN = __powf(d, (float)u);
    const float d16   = __powf(d, 16.0f);

    // Incoming carry for this segment: A_g = sum_{g'<g} L[g'] * d^(SEG*(g-1-g'))
    float carry = 0.0f;
    if (g > 0) {
        const float dSEG = __powf(d, (float)SEG);
        const float* Lrow = L + (size_t)(rowBase + u) * G;
        for (int gp = 0; gp < g; ++gp)
            carry = carry * dSEG + Lrow[gp];      // Horner
    }

    const float* xrow = x + (size_t)(rowBase + u) * (size_t)S;
    const int cbBeg = g * SEG, cbEnd = cbBeg + SEG;

    for (int cb = cbBeg; cb < cbEnd; cb += 16) {
        v2f a[4], b[4];
        float partial = 0.0f;
        #pragma unroll
        for (int s = 0; s < 4; ++s) {
            const int col = cb + 4*s + 2*h;
            a[s] = *(const v2f*)(xrow + col);         // A: row u, K=col..col+1
            const v2f vv  = *(const v2f*)(w  + col);  // 8KB vectors: cache-hot
            const v2f v2v = *(const v2f*)(wd + col);
            // B[j,u] = v[j]*d^(u-j) (u>j) | v2[j] (u==j) | 0
            b[s].x = vv.x * tri[s][0] + (dg[s][0] ? v2v.x : 0.0f);
            b[s].y = vv.y * tri[s][1] + (dg[s][1] ? v2v.y : 0.0f);
            partial += a[s].x*vv.x*dw[s][0] + a[s].y*vv.y*dw[s][1];
        }

        v8f D = {};
        #pragma unroll
        for (int s = 0; s < 4; ++s) {
            D = __builtin_amdgcn_wmma_f32_16x16x4_f32(
                    false, a[s], false, b[s], (short)0, D, false, false);
        }

        const float tot  = partial + __shfl_xor(partial, 16, 32);
        const float bval = bias[cb + u];

        #pragma unroll
        for (int v = 0; v < 8; ++v) {
            const float cv = __shfl(carry, v + 8*h, 32);   // carry[row v+8h]
            D[v] += fmaf(cv, dpowN, bval);
        }
        #pragma unroll
        for (int v = 0; v < 8; ++v) {
            out[(size_t)(rowBase + v + 8*h) * (size_t)S + (cb + u)] = D[v];
        }

        carry = d16 * carry + tot;                  // advance scan state
    }
}

extern "C" void kernel_launch(void* const* d_in, const int* in_sizes, int n_in,
                              void* d_out, int out_size, void* d_ws, size_t ws_size,
                              hipStream_t stream) {
    const float* x    = (const float*)d_in[0];  // (B,E,S) fp32
    const float* w    = (const float*)d_in[1];  // (1,S)
    const float* wd   = (const float*)d_in[2];  // (1,S)
    const float* bias = (const float*)d_in[3];  // (S,)
    const float* dec  = (const float*)d_in[4];  // (2,1)
    float* out = (float*)d_out;

    const int S = in_sizes[1];            // 2048
    const int R = in_sizes[0] / S;        // B*E = 16384 rows
    const int tiles = R / 16;             // 1024 row tiles

    // Segment count: 4 segments -> 4096 waves (~3+/SIMD on ~1280 SIMD32s).
    int G = 4;
    if (S % (16 * G) != 0 || ws_size < (size_t)R * G * sizeof(float)) G = 1;
    const int SEG = S / G;
    float* L = (float*)d_ws;

    const int wavesPerBlock = 256 / 32;   // 8
    const int tasks = tiles * G;
    const int blocks = (tasks + wavesPerBlock - 1) / wavesPerBlock;

    if (G > 1)
        seg_carry_kernel<<<blocks, 256, 0, stream>>>(x, w, dec, L, S, R, G, SEG);
    seg_scan_kernel<<<blocks, 256, 0, stream>>>(x, w, wd, bias, dec, L, out,
                                                S, R, G, SEG);
}